// GlobalShapeStatistics_40338332844250
// MI455X (gfx1250) — compile-verified
//
#include <hip/hip_runtime.h>
#include <stdint.h>

// Problem constants (match reference)
#define B_DIM   64
#define H_DIM   1024
#define W_DIMM  1024
#define OUT_DIM 256
#define ELEMS   (H_DIM * W_DIMM)      // 1,048,576 per image
#define QUADS   (ELEMS / 4)           // 262,144 float4 per image
#define RED_BLOCKS  128
#define RED_THREADS 256
#define TPI     (RED_BLOCKS * RED_THREADS)  // 32768 threads per image
#define QPT     (QUADS / TPI)               // 8 float4 per thread

typedef float v2f __attribute__((ext_vector_type(2)));
typedef float v8f __attribute__((ext_vector_type(8)));

// Workspace layout (u32 stats per image):
// [0]=area  [1]=sum_r  [2]=sum_c  [3]=min_r  [4]=max_r  [5]=min_c  [6]=max_c  [7]=pad
__global__ void gss_init_stats(unsigned* __restrict__ stats) {
    unsigned* s = stats + threadIdx.x * 8;
    s[0] = 0u; s[1] = 0u; s[2] = 0u;
    s[3] = 0xFFFFFFFFu; s[4] = 0u;
    s[5] = 0xFFFFFFFFu; s[6] = 0u; s[7] = 0u;
}

// Bandwidth-bound mask reduction: float4 loads, u32 accumulation (exact),
// wave32 shuffle reduction -> LDS -> 7 global atomics per block.
__global__ __launch_bounds__(RED_THREADS)
void gss_reduce_mask(const float4* __restrict__ mask4, unsigned* __restrict__ stats) {
    const int b = blockIdx.y;
    const unsigned tid_img = blockIdx.x * RED_THREADS + threadIdx.x;
    const float4* img = mask4 + (size_t)b * QUADS;

    unsigned area = 0u, sr = 0u, sc = 0u;
    unsigned minr = 0xFFFFFFFFu, maxr = 0u, minc = 0xFFFFFFFFu, maxc = 0u;

#pragma unroll
    for (int i = 0; i < QPT; ++i) {
        const unsigned q  = tid_img + (unsigned)i * TPI;
        const float4   v  = img[q];
        const unsigned r  = q >> 8;              // 256 float4 per row
        const unsigned c0 = (q & 255u) << 2;

        const unsigned m0 = (v.x != 0.0f) ? 1u : 0u;
        const unsigned m1 = (v.y != 0.0f) ? 1u : 0u;
        const unsigned m2 = (v.z != 0.0f) ? 1u : 0u;
        const unsigned m3 = (v.w != 0.0f) ? 1u : 0u;
        const unsigned nz = m0 + m1 + m2 + m3;

        area += nz;
        sr   += nz * r;
        sc   += nz * c0 + (m1 + 2u * m2 + 3u * m3);

        // bbox: row bounds once per quad, col bounds via first/last set lane
        const unsigned lo = m0 ? 0u : (m1 ? 1u : (m2 ? 2u : 3u));
        const unsigned hi = m3 ? 3u : (m2 ? 2u : (m1 ? 1u : 0u));
        const unsigned cmin = c0 + lo, cmax = c0 + hi;
        minr = (nz && r    < minr) ? r    : minr;
        maxr = (nz && r    > maxr) ? r    : maxr;
        minc = (nz && cmin < minc) ? cmin : minc;
        maxc = (nz && cmax > maxc) ? cmax : maxc;
    }

    // wave32 butterfly reduction
#pragma unroll
    for (int off = 16; off > 0; off >>= 1) {
        area += (unsigned)__shfl_xor((int)area, off, 32);
        sr   += (unsigned)__shfl_xor((int)sr,   off, 32);
        sc   += (unsigned)__shfl_xor((int)sc,   off, 32);
        unsigned t;
        t = (unsigned)__shfl_xor((int)minr, off, 32); minr = t < minr ? t : minr;
        t = (unsigned)__shfl_xor((int)maxr, off, 32); maxr = t > maxr ? t : maxr;
        t = (unsigned)__shfl_xor((int)minc, off, 32); minc = t < minc ? t : minc;
        t = (unsigned)__shfl_xor((int)maxc, off, 32); maxc = t > maxc ? t : maxc;
    }

    __shared__ unsigned s_a[8], s_sr[8], s_sc[8], s_mnr[8], s_mxr[8], s_mnc[8], s_mxc[8];
    const int wid  = threadIdx.x >> 5;
    const int lane = threadIdx.x & 31;
    if (lane == 0) {
        s_a[wid] = area; s_sr[wid] = sr; s_sc[wid] = sc;
        s_mnr[wid] = minr; s_mxr[wid] = maxr; s_mnc[wid] = minc; s_mxc[wid] = maxc;
    }
    __syncthreads();
    if (threadIdx.x == 0) {
#pragma unroll
        for (int w = 1; w < RED_THREADS / 32; ++w) {
            area += s_a[w]; sr += s_sr[w]; sc += s_sc[w];
            minr = s_mnr[w] < minr ? s_mnr[w] : minr;
            maxr = s_mxr[w] > maxr ? s_mxr[w] : maxr;
            minc = s_mnc[w] < minc ? s_mnc[w] : minc;
            maxc = s_mxc[w] > maxc ? s_mxc[w] : maxc;
        }
        unsigned* s = stats + b * 8;
        atomicAdd(&s[0], area);
        atomicAdd(&s[1], sr);
        atomicAdd(&s[2], sc);
        atomicMin(&s[3], minr);
        atomicMax(&s[4], maxr);
        atomicMin(&s[5], minc);
        atomicMax(&s[6], maxc);
    }
}

// Integer stats -> f32 features, padded to K=8 with zeros.
__global__ void gss_feats(const unsigned* __restrict__ stats, float* __restrict__ feats) {
    const unsigned* s = stats + threadIdx.x * 8;
    const float area   = (float)s[0];
    const float safe   = fmaxf(area, 1.0f);
    const float cr     = (float)s[1] / safe;
    const float cc     = (float)s[2] / safe;
    const float height = (float)s[4] - (float)s[3];
    const float width  = (float)s[6] - (float)s[5];
    const float hw     = (float)ELEMS;

    float f0 = cr / (float)H_DIM;
    float f1 = cc / (float)W_DIMM;
    float f2 = height / (float)H_DIM;
    float f3 = width / (float)W_DIMM;
    float f4 = area / hw;
    float f5 = height * width / hw;
    if (s[0] == 0u) { f0 = f1 = f2 = f3 = f4 = f5 = 0.0f; }

    float* o = feats + threadIdx.x * 8;
    o[0] = f0; o[1] = f1; o[2] = f2; o[3] = f3; o[4] = f4; o[5] = f5;
    o[6] = 0.0f; o[7] = 0.0f;
}

// out[64,256] = feats[64,8(pad)] @ W^T + b  via V_WMMA_F32_16X16X4_F32.
// One wave per 16x16 output tile; K=8 -> two chained WMMAs.
// A layout (16x4 f32): lanes 0-15 hold {K0,K1} for M=lane; lanes 16-31 hold {K2,K3}.
// B layout (4x16 f32): lanes 0-15 hold {K0,K1} for N=lane; lanes 16-31 hold {K2,K3}.
// C/D layout (16x16 f32): lane<16 -> N=lane, VGPR v = M=v; lane>=16 -> M=8+v.
__global__ void gss_wmma_gemm(const float* __restrict__ feats,
                              const float* __restrict__ Wm,
                              const float* __restrict__ bias,
                              float* __restrict__ out) {
    const int lane = threadIdx.x;          // 0..31
    const int tile = blockIdx.x;           // 0..63
    const int tm   = (tile >> 4) << 4;     // 0,16,32,48   (M tiles: 64/16)
    const int tn   = (tile & 15) << 4;     // 0..240       (N tiles: 256/16)
    const int hi   = lane >> 4;            // 0 or 1
    const int l    = lane & 15;

    // A fragments (feats row-major, stride 8, padded with zeros at k=6,7)
    const int m  = tm + l;
    const int ka = hi * 2;
    v2f a0, a1;
    a0.x = feats[m * 8 + ka];
    a0.y = feats[m * 8 + ka + 1];
    a1.x = feats[m * 8 + ka + 4];
    a1.y = feats[m * 8 + ka + 5];

    // B fragments: B[k][n] = W[n][k], W row-major (256,6)
    const int n  = tn + l;
    const int kb = hi * 2;
    v2f b0, b1;
    b0.x = Wm[n * 6 + kb];
    b0.y = Wm[n * 6 + kb + 1];
    b1.x = (kb + 4 < 6) ? Wm[n * 6 + kb + 4] : 0.0f;
    b1.y = (kb + 5 < 6) ? Wm[n * 6 + kb + 5] : 0.0f;

    v8f c = {0.f, 0.f, 0.f, 0.f, 0.f, 0.f, 0.f, 0.f};
    c = __builtin_amdgcn_wmma_f32_16x16x4_f32(false, a0, false, b0, (short)0, c, false, false);
    c = __builtin_amdgcn_wmma_f32_16x16x4_f32(false, a1, false, b1, (short)0, c, false, false);

    const float bb = bias[n];
    const int mbase = tm + hi * 8;
#pragma unroll
    for (int v = 0; v < 8; ++v) {
        out[(mbase + v) * OUT_DIM + n] = c[v] + bb;
    }
}

extern "C" void kernel_launch(void* const* d_in, const int* in_sizes, int n_in,
                              void* d_out, int out_size, void* d_ws, size_t ws_size,
                              hipStream_t stream) {
    const float* mask = (const float*)d_in[0];   // (64,1,1024,1024) f32
    const float* Wm   = (const float*)d_in[1];   // (256,6) f32
    const float* bias = (const float*)d_in[2];   // (256,) f32
    float* out = (float*)d_out;                  // (64,256) f32

    unsigned* stats = (unsigned*)d_ws;                              // 64*8 u32
    float*    feats = (float*)((char*)d_ws + B_DIM * 8 * sizeof(unsigned)); // 64*8 f32

    gss_init_stats<<<1, B_DIM, 0, stream>>>(stats);
    gss_reduce_mask<<<dim3(RED_BLOCKS, B_DIM), RED_THREADS, 0, stream>>>(
        (const float4*)mask, stats);
    gss_feats<<<1, B_DIM, 0, stream>>>(stats, feats);
    gss_wmma_gemm<<<64, 32, 0, stream>>>(feats, Wm, bias, out);
}